// GraphConv_25847113187704
// MI455X (gfx1250) — compile-verified
//
#include <hip/hip_runtime.h>

typedef __attribute__((ext_vector_type(2))) float v2f;
typedef __attribute__((ext_vector_type(8))) float v8f;

#define N_NODES 100000
#define N_EDGES 1600000
#define IN_F    128
#define OUT_F   32
#define N_TILES (N_NODES / 16)   // 6250, exact

// ---------------- kernel 0: zero out + degree buffers -----------------------
__global__ void gc_zero(unsigned* __restrict__ outdeg, unsigned* __restrict__ indeg,
                        float* __restrict__ out) {
    int t = blockIdx.x * blockDim.x + threadIdx.x;
    if (t < N_NODES * OUT_F) out[t] = 0.0f;
    if (t < N_NODES) { outdeg[t] = 0u; indeg[t] = 0u; }
}

// ---------------- kernel 1: degree histogram --------------------------------
__global__ void gc_degrees(const int* __restrict__ src, const int* __restrict__ dst,
                           unsigned* __restrict__ outdeg, unsigned* __restrict__ indeg) {
    int e = blockIdx.x * blockDim.x + threadIdx.x;
    if (e < N_EDGES) {
        atomicAdd(&outdeg[src[e]], 1u);
        atomicAdd(&indeg[dst[e]], 1u);
    }
}

// ---------------- kernel 2: h = (feat * outdeg^-1/2) @ W  via f32 WMMA ------
// One wave (32 lanes) computes a 16x32 output tile = two 16x16 WMMA tiles,
// accumulating 32 K-steps of V_WMMA_F32_16X16X4_F32.
__global__ void __launch_bounds__(256)
gc_gemm_wmma(const float* __restrict__ feat, const float* __restrict__ W,
             const unsigned* __restrict__ outdeg, float* __restrict__ h) {
    __shared__ float Wlds[IN_F * OUT_F];      // 16 KB: whole weight matrix
    for (int i = threadIdx.x; i < IN_F * OUT_F; i += 256) Wlds[i] = W[i];
    __syncthreads();

    const int wave = threadIdx.x >> 5;
    const int lane = threadIdx.x & 31;
    const int tile = blockIdx.x * 8 + wave;
    if (tile >= N_TILES) return;              // wave-uniform: EXEC stays all-1s

    const int half = lane >> 4;               // 0: lanes 0-15, 1: lanes 16-31
    const int lm   = lane & 15;               // M (for A/D) or N (for B/D)
    const int row0 = tile * 16;
    const int row  = row0 + lm;

    // left norm: out_deg^-1/2 (clip >= 1)
    const float s = 1.0f / sqrtf(fmaxf((float)outdeg[row], 1.0f));
    // A-frag base: 16x4 f32 A layout -> lane holds A[M][k0 + 2*half + {0,1}]
    const float* __restrict__ arow = feat + (size_t)row * IN_F + 2 * half;

    v8f c0 = {0.f,0.f,0.f,0.f,0.f,0.f,0.f,0.f};
    v8f c1 = {0.f,0.f,0.f,0.f,0.f,0.f,0.f,0.f};

#pragma unroll
    for (int ks = 0; ks < IN_F / 4; ++ks) {
        const int k0 = ks * 4;
        v2f a;                                 // A 16x4: K = k0 + 2*half + v
        a.x = arow[k0]     * s;
        a.y = arow[k0 + 1] * s;

        const int kb = k0 + 2 * half;          // B 4x16: K = k0 + 2*half + v, N = lm
        v2f b0, b1;
        b0.x = Wlds[kb       * OUT_F + lm];
        b0.y = Wlds[(kb + 1) * OUT_F + lm];
        b1.x = Wlds[kb       * OUT_F + 16 + lm];
        b1.y = Wlds[(kb + 1) * OUT_F + 16 + lm];

        c0 = __builtin_amdgcn_wmma_f32_16x16x4_f32(false, a, false, b0,
                                                   (short)0, c0, false, false);
        c1 = __builtin_amdgcn_wmma_f32_16x16x4_f32(false, a, false, b1,
                                                   (short)0, c1, false, false);
    }

    // D layout: VGPR r, lane-half hf -> M = r + 8*hf, N = lm (+16 for tile 1)
    float* __restrict__ hp = h + (size_t)row0 * OUT_F;
#pragma unroll
    for (int r = 0; r < 8; ++r) {
        const int m = r + 8 * half;
        hp[m * OUT_F + lm]      = c0[r];
        hp[m * OUT_F + 16 + lm] = c1[r];
    }
}

// ---------------- kernel 3: edge scatter  out[dst] += h[src] ----------------
// 32 consecutive threads (one wave) handle one edge: contiguous 128B gather
// of h[src] and a contiguous 128B burst of global_atomic_add_f32 to out[dst].
__global__ void gc_scatter(const int* __restrict__ src, const int* __restrict__ dst,
                           const float* __restrict__ h, float* __restrict__ out) {
    int t = blockIdx.x * blockDim.x + threadIdx.x;   // < 51.2M, fits int
    int e = t >> 5;
    int j = t & 31;
    if (e < N_EDGES) {
        int sN = src[e];
        int dN = dst[e];
        atomicAdd(&out[(size_t)dN * OUT_F + j], h[(size_t)sN * OUT_F + j]);
    }
}

// ---------------- kernel 4: right norm  out *= indeg^-1/2 -------------------
__global__ void gc_scale(const unsigned* __restrict__ indeg, float* __restrict__ out) {
    int t = blockIdx.x * blockDim.x + threadIdx.x;
    if (t < N_NODES * OUT_F) {
        int i = t >> 5;                                // OUT_F == 32
        out[t] *= 1.0f / sqrtf(fmaxf((float)indeg[i], 1.0f));
    }
}

extern "C" void kernel_launch(void* const* d_in, const int* in_sizes, int n_in,
                              void* d_out, int out_size, void* d_ws, size_t ws_size,
                              hipStream_t stream) {
    const float* feat   = (const float*)d_in[0];   // [100000,128]
    const int*   src    = (const int*)  d_in[1];   // [1600000]
    const int*   dst    = (const int*)  d_in[2];   // [1600000]
    const float* weight = (const float*)d_in[3];   // [128,32]
    float*       out    = (float*)d_out;           // [100000,32]

    // workspace layout: outdeg | indeg | h   (~13.6 MB total)
    unsigned* outdeg = (unsigned*)d_ws;
    unsigned* indeg  = outdeg + N_NODES;
    float*    h      = (float*)(indeg + N_NODES);

    gc_zero     <<<(N_NODES * OUT_F + 255) / 256, 256, 0, stream>>>(outdeg, indeg, out);
    gc_degrees  <<<(N_EDGES + 255) / 256,        256, 0, stream>>>(src, dst, outdeg, indeg);
    gc_gemm_wmma<<<(N_TILES + 7) / 8,            256, 0, stream>>>(feat, weight, outdeg, h);
    gc_scatter  <<<(N_EDGES * 32) / 256,         256, 0, stream>>>(src, dst, h, out);
    gc_scale    <<<(N_NODES * OUT_F + 255) / 256,256, 0, stream>>>(indeg, out);
}